// MultiLoraLinear_83623013253472
// MI455X (gfx1250) — compile-verified
//
#include <hip/hip_runtime.h>
#include <hip/hip_bf16.h>

typedef float v2f __attribute__((ext_vector_type(2)));
typedef float v8f __attribute__((ext_vector_type(8)));
typedef int   v4i __attribute__((ext_vector_type(4)));

#define IN_DIM   4096
#define OUT_DIM  4096
#define S_DIM    2048
#define B_DIM    8

#define TILE_M   128   // S rows per block
#define TILE_N   128   // OUT cols per block
#define KT       32    // K chunk staged in LDS
#define LDSS     36    // LDS row stride in floats (144B: 16B-aligned, conflict-free)

// ---- CDNA5 async global->LDS copy (ASYNCcnt path, no VGPR staging) ----------
__device__ __forceinline__ void async_load_b128(const float* gptr, float* lptr) {
#if __has_builtin(__builtin_amdgcn_global_load_async_to_lds_b128)
    __builtin_amdgcn_global_load_async_to_lds_b128(
        (__attribute__((address_space(1))) v4i*)(v4i*)const_cast<float*>(gptr),
        (__attribute__((address_space(3))) v4i*)(v4i*)lptr,
        /*offset=*/0, /*cpol=*/0);
#else
    const unsigned int lds_off = (unsigned int)(unsigned long long)lptr;
    asm volatile("global_load_async_to_lds_b128 %0, %1, off"
                 :: "v"(lds_off), "v"(gptr) : "memory");
#endif
}

template <int N>
__device__ __forceinline__ void wait_asynccnt() {
#if __has_builtin(__builtin_amdgcn_s_wait_asynccnt)
    __builtin_amdgcn_s_wait_asynccnt(N);
#else
    asm volatile("s_wait_asynccnt %0" :: "i"(N) : "memory");
#endif
}

// Issue one 128x32 fp32 tile pair (x and W) as 8 async B128 transfers/thread.
__device__ __forceinline__ void stage_tiles_async(const float* xb, const float* wb,
                                                  int k0, float* Asb, float* Bsb,
                                                  int tid) {
    #pragma unroll
    for (int p = 0; p < 4; ++p) {
        const int e4  = tid + p * 256;   // float4 index within 128x8 grid
        const int row = e4 >> 3;         // 8 float4 per 32-float row
        const int col = (e4 & 7) << 2;
        const size_t goff = (size_t)row * IN_DIM + k0 + col;
        const int    loff = row * LDSS + col;
        async_load_b128(xb + goff, Asb + loff);
        async_load_b128(wb + goff, Bsb + loff);
    }
}

// out[b, m0+m, n0+n] = sum_k x[b, m0+m, k] * W[aid[b], n0+n, k]
__launch_bounds__(256)
__global__ void multilora_wmma_f32_kernel(const float* __restrict__ x,
                                          const int*   __restrict__ adapter_ids,
                                          const float* __restrict__ weight,
                                          float*       __restrict__ out)
{
    __shared__ float As[2][TILE_M * LDSS];   // x tile  [m][k], double-buffered
    __shared__ float Bs[2][TILE_N * LDSS];   // W tile  [n][k], double-buffered

    const int b   = blockIdx.z;
    const int m0  = blockIdx.y * TILE_M;  // S offset
    const int n0  = blockIdx.x * TILE_N;  // OUT offset
    const int aid = adapter_ids[b];

    const float* xb = x      + (size_t)b   * S_DIM   * IN_DIM + (size_t)m0 * IN_DIM;
    const float* wb = weight + (size_t)aid * OUT_DIM * IN_DIM + (size_t)n0 * IN_DIM;

    const int tid   = threadIdx.x;
    const int lane  = tid & 31;
    const int wid   = tid >> 5;
    const int waveM = wid >> 1;            // 0..3  -> 32-row strip
    const int waveN = wid & 1;             // 0..1  -> 64-col strip
    const int lrow  = lane & 15;           // row/col within 16x16 fragment
    const int khalf = (lane >> 4) << 1;    // lanes 16-31 hold K+2,K+3

    v8f acc[2][4] = {};                    // 32x64 per wave as 2x4 tiles of 16x16

    // Prologue: fill buffer 0.
    stage_tiles_async(xb, wb, 0, As[0], Bs[0], tid);

    for (int k0 = 0; k0 < IN_DIM; k0 += KT) {
        const int cur = (k0 >> 5) & 1;     // KT == 32

        if (k0 + KT < IN_DIM) {
            // Overlap: issue next chunk's transfers, then wait only for the
            // 8 older (current-buffer) transfers. Async loads retire in order.
            stage_tiles_async(xb, wb, k0 + KT, As[cur ^ 1], Bs[cur ^ 1], tid);
            wait_asynccnt<8>();
        } else {
            wait_asynccnt<0>();
        }
        __syncthreads();                   // all waves' writes to `cur` visible

        const float* Asb = As[cur];
        const float* Bsb = Bs[cur];

        #pragma unroll
        for (int kk = 0; kk < KT; kk += 4) {
            v2f afrag[2];
            v2f bfrag[4];
            // A fragment: 16x4 f32. lanes 0-15 => (M=lrow, K=kk..kk+1),
            //             lanes 16-31 => (M=lrow, K=kk+2..kk+3)
            #pragma unroll
            for (int mt = 0; mt < 2; ++mt) {
                const float* p = Asb + (waveM * 32 + mt * 16 + lrow) * LDSS + kk + khalf;
                afrag[mt] = *(const v2f*)p;
            }
            // B fragment: 4x16 f32, lane = N column, same K split.
            #pragma unroll
            for (int nt = 0; nt < 4; ++nt) {
                const float* p = Bsb + (waveN * 64 + nt * 16 + lrow) * LDSS + kk + khalf;
                bfrag[nt] = *(const v2f*)p;
            }
            #pragma unroll
            for (int mt = 0; mt < 2; ++mt) {
                #pragma unroll
                for (int nt = 0; nt < 4; ++nt) {
                    acc[mt][nt] = __builtin_amdgcn_wmma_f32_16x16x4_f32(
                        /*neg_a=*/false, afrag[mt],
                        /*neg_b=*/false, bfrag[nt],
                        /*c_mod=*/(short)0, acc[mt][nt],
                        /*reuse_a=*/false, /*reuse_b=*/false);
                }
            }
        }
        __syncthreads();                   // reads done before next issue overwrites
    }

    // C/D layout: VGPR r, lanes 0-15 => (M=r, N=lane); lanes 16-31 => (M=r+8, N=lane-16)
    float* ob = out + (size_t)b * S_DIM * OUT_DIM + (size_t)m0 * OUT_DIM + n0;
    const int mlane = (lane >> 4) << 3;    // 0 or 8
    #pragma unroll
    for (int mt = 0; mt < 2; ++mt) {
        #pragma unroll
        for (int nt = 0; nt < 4; ++nt) {
            #pragma unroll
            for (int r = 0; r < 8; ++r) {
                const int m = waveM * 32 + mt * 16 + mlane + r;
                const int n = waveN * 64 + nt * 16 + lrow;
                ob[(size_t)m * OUT_DIM + n] = acc[mt][nt][r];
            }
        }
    }
}

extern "C" void kernel_launch(void* const* d_in, const int* in_sizes, int n_in,
                              void* d_out, int out_size, void* d_ws, size_t ws_size,
                              hipStream_t stream) {
    const float* x   = (const float*)d_in[0];
    const int*   aid = (const int*)  d_in[1];
    const float* w   = (const float*)d_in[2];
    float*       out = (float*)d_out;

    dim3 grid(OUT_DIM / TILE_N, S_DIM / TILE_M, B_DIM);
    dim3 block(256, 1, 1);
    multilora_wmma_f32_kernel<<<grid, block, 0, stream>>>(x, aid, w, out);
}